// Potential_41412074668738
// MI455X (gfx1250) — compile-verified
//
#include <hip/hip_runtime.h>

typedef float v2f __attribute__((ext_vector_type(2)));
typedef float v8f __attribute__((ext_vector_type(8)));

#define EPS_F    0.001f
#define INV_EPS  1000.0f
#define LOG2E_F  1.4426950408889634f
#define LN2_F    0.6931471805599453f
#define NEG_BIG  (-3.4e38f)

// ---------------------------------------------------------------------------
// Kernel 1: L = logsumexp(logprob) over M elements -> ws[0]
// ---------------------------------------------------------------------------
__global__ void lse_reduce_kernel(const float* __restrict__ lp, int m,
                                  float* __restrict__ ws) {
    __shared__ float red[256];
    const int t = threadIdx.x;

    float mx = NEG_BIG;
    for (int i = t; i < m; i += 256) mx = fmaxf(mx, lp[i]);
    red[t] = mx;
    __syncthreads();
    for (int s = 128; s > 0; s >>= 1) {
        if (t < s) red[t] = fmaxf(red[t], red[t + s]);
        __syncthreads();
    }
    const float gmax = red[0];
    __syncthreads();

    float sum = 0.0f;
    for (int i = t; i < m; i += 256) sum += __expf(lp[i] - gmax);
    red[t] = sum;
    __syncthreads();
    for (int s = 128; s > 0; s >>= 1) {
        if (t < s) red[t] += red[t + s];
        __syncthreads();
    }
    if (t == 0) ws[0] = gmax + __logf(red[0]);
}

// ---------------------------------------------------------------------------
// Kernel 2: out[n] = x[n]^2/2 - eps*ln2*( mx2[n] + log2 sum_m 2^(t2 - mx2) )
//   t2[n,m] = log2e*( c[m] + x[n]*p[m] ), generated by V_WMMA_F32_16X16X4_F32:
//     A (16x4) rows = (x, 1, *, *)   [K=2,3 exactly zero in lanes 16-31]
//     B (4x16) cols = (p2[m], c2[m], dup, dup)^T  [K=2,3 garbage x A's zeros = 0]
//   Pass 2 folds -rowmax into the WMMA C accumulator, so the per-element work
//   is exactly one v_exp_f32 + one v_add_f32.
//   Block = 256 threads = 8 wave32s; wave owns 16 rows; all M (p2,c2) pairs
//   staged in dynamic LDS (M*8 bytes = 128KB, fits 320KB/WGP).
// ---------------------------------------------------------------------------
__global__ void potential_kernel(const float* __restrict__ x,
                                 const float* __restrict__ pos,
                                 const float* __restrict__ lp,
                                 const float* __restrict__ pot,
                                 const float* __restrict__ ws,
                                 float* __restrict__ out,
                                 int n, int m) {
    extern __shared__ float smem[];           // float2 (p2, c2) per support pt
    float2* sPC = (float2*)smem;

    const float L = ws[0];

    // Stage: p2 = log2e * pos/eps ; c2 = log2e * (lp - L + (pot - pos^2/2)/eps)
    for (int i = threadIdx.x; i < m; i += blockDim.x) {
        const float p = pos[i];
        const float c = lp[i] - L + (pot[i] - 0.5f * p * p) * INV_EPS;
        sPC[i] = make_float2(p * (INV_EPS * LOG2E_F), c * LOG2E_F);
    }
    __syncthreads();

    const int wave = threadIdx.x >> 5;
    const int lane = threadIdx.x & 31;
    const int half = lane >> 4;               // 0: rows 0-7, 1: rows 8-15
    const int l16  = lane & 15;
    const int rowBase = blockIdx.x * 128 + wave * 16;
    const float msk = (half == 0) ? 1.0f : 0.0f;

    // A fragment: lanes 0-15 (K=0,1) -> (x[row], 1); lanes 16-31 (K=2,3) -> 0.
    // A's zero K=2,3 columns null out whatever lanes 16-31 supply for B.
    v2f a;
    a.x = x[rowBase + l16] * msk;
    a.y = msk;

    const v8f cz = {0.f, 0.f, 0.f, 0.f, 0.f, 0.f, 0.f, 0.f};

    // ---------------- Pass 1: row maxima of t2 ----------------
    v8f mx = {NEG_BIG, NEG_BIG, NEG_BIG, NEG_BIG,
              NEG_BIG, NEG_BIG, NEG_BIG, NEG_BIG};
#pragma unroll 8
    for (int mb = 0; mb < m; mb += 16) {
        const float2 pc = sPC[mb + l16];      // lanes 16-31: dup read (harmless)
        v2f b; b.x = pc.x; b.y = pc.y;        // no masking needed
        v8f d = __builtin_amdgcn_wmma_f32_16x16x4_f32(
            false, a, false, b, (short)0, cz, false, false);
#pragma unroll
        for (int v = 0; v < 8; ++v) mx[v] = fmaxf(mx[v], d[v]);
    }
    // Combine maxima across the 16 lanes sharing each row (masks stay in-half)
#pragma unroll
    for (int sh = 1; sh <= 8; sh <<= 1) {
#pragma unroll
        for (int v = 0; v < 8; ++v)
            mx[v] = fmaxf(mx[v], __shfl_xor(mx[v], sh, 32));
    }

    // Negated row max, replicated across each row's lanes in C/D layout:
    // WMMA computes t2 - rowmax directly.
    v8f negmx;
#pragma unroll
    for (int v = 0; v < 8; ++v) negmx[v] = -mx[v];

    // ---------------- Pass 2: sum of 2^(t2 - rowmax) ----------------
    v8f sum = cz;
#pragma unroll 8
    for (int mb = 0; mb < m; mb += 16) {
        const float2 pc = sPC[mb + l16];
        v2f b; b.x = pc.x; b.y = pc.y;
        v8f d = __builtin_amdgcn_wmma_f32_16x16x4_f32(
            false, a, false, b, (short)0, negmx, false, false);
#pragma unroll
        for (int v = 0; v < 8; ++v) sum[v] += __builtin_amdgcn_exp2f(d[v]);
    }
    // Shared max per row -> cross-lane combine is a plain sum
#pragma unroll
    for (int sh = 1; sh <= 8; sh <<= 1) {
#pragma unroll
        for (int v = 0; v < 8; ++v) sum[v] += __shfl_xor(sum[v], sh, 32);
    }

    // ---------------- Writeback ----------------
    // Lane 0 writes rows rowBase+0..7 (slots 0..7); lane 16 writes rowBase+8..15.
    if (l16 == 0) {
        const int rb = rowBase + half * 8;
#pragma unroll
        for (int v = 0; v < 8; ++v) {
            const float xr = x[rb + v];
            // ln(sum_m e^t) = ln2 * (mx2 + log2(sum2))
            out[rb + v] = 0.5f * xr * xr
                        - (EPS_F * LN2_F) * (mx[v] + __builtin_amdgcn_logf(sum[v]));
        }
    }
}

// ---------------------------------------------------------------------------
extern "C" void kernel_launch(void* const* d_in, const int* in_sizes, int n_in,
                              void* d_out, int out_size, void* d_ws, size_t ws_size,
                              hipStream_t stream) {
    const float* x   = (const float*)d_in[0];
    const float* pos = (const float*)d_in[1];
    const float* lp  = (const float*)d_in[2];
    const float* pot = (const float*)d_in[3];
    float* out = (float*)d_out;
    float* ws  = (float*)d_ws;

    const int n = in_sizes[0];   // 16384 samples
    const int m = in_sizes[2];   // 16384 support points

    lse_reduce_kernel<<<1, 256, 0, stream>>>(lp, m, ws);

    const int rows_per_block = 128;            // 8 wave32s * 16 rows
    const int grid = (n + rows_per_block - 1) / rows_per_block;
    const size_t lds_bytes = (size_t)m * 2 * sizeof(float);  // 128 KB
    potential_kernel<<<grid, 256, lds_bytes, stream>>>(x, pos, lp, pot, ws,
                                                       out, n, m);
}